// Monarch_52570399703167
// MI455X (gfx1250) — compile-verified
//
#include <hip/hip_runtime.h>

// Monarch (block-diag matmul -> permute -> block-diag matmul -> permute -> +bias)
// Two-pass fp32 WMMA implementation for gfx1250 (MI455X).
//   SIZE=8192, NB=64 blocks, BS=128, NTOK=4096.
// Pass 1: h1 = blockdiag(w1) @ x, stored PERMUTED (inter[b, c*64+k]) so pass 2
//         reads contiguously. Intermediate (134MB) lives in the 192MB L2.
// Pass 2: out = permute2(blockdiag(w2) @ inter) + bias, permute folded into
//         scattered stores that merge in the write-back L2.

#define SIZE 8192
#define NB   64
#define BS   128
#define TM   64      // tokens per workgroup
#define KH   64      // K-half staged in LDS per iteration
#define LDX  68      // padded LDS row stride in floats (16B-aligned rows, bank-conflict free)

typedef float v2f __attribute__((ext_vector_type(2)));
typedef float v8f __attribute__((ext_vector_type(8)));

// PASS==1: src=x,     w=w1, dst=inter (permute-1 on store)
// PASS==2: src=inter, w=w2, dst=out   (permute-2 + bias on store)
template <int PASS>
__global__ __launch_bounds__(256) void monarch_pass(
    const float* __restrict__ src, const float* __restrict__ w,
    const float* __restrict__ bias, float* __restrict__ dst) {
  __shared__ float Xs[TM][LDX];    // token panel half   (64 x 64 used)
  __shared__ float Wst[BS][LDX];   // weight block half, TRANSPOSED: Wst[c][d]

  const int tid  = threadIdx.x;
  const int lane = tid & 31;       // wave32
  const int wv   = tid >> 5;       // 8 waves
  const int b0   = blockIdx.x * TM;
  const int kblk = blockIdx.y;     // block index (k for pass1, k2 for pass2)

  const float* __restrict__ wblk = w + (size_t)kblk * BS * BS;

  // WMMA f32 16x16x4 fragment mapping (wave32):
  //  A: lane m = lane&15, holds K = 2*(lane>>4) + {0,1}   -> v2f
  //  B: lane n = lane&15, holds K = 2*(lane>>4) + {0,1}   -> v2f
  //  C/D: vgpr r -> M = r + 8*(lane>>4), N = lane&15      -> v8f
  const int mrow  = (wv & 3) * 16 + (lane & 15);
  const int khalf = (lane >> 4) << 1;        // 0 or 2
  const int nbase = (wv >> 2) * 64;          // this wave's 4 N-tiles start
  const int ncol  = lane & 15;

  v8f acc[4] = {};                           // 4 N-tiles of 16x16 f32

  #pragma unroll
  for (int kh = 0; kh < 2; ++kh) {
    __syncthreads();
    // ---- stage token panel half: rows 0..63, cols kh*64..kh*64+63 (coalesced b128)
    #pragma unroll
    for (int i = 0; i < 4; ++i) {
      int f  = tid + i * 256;
      int r  = f >> 4;
      int c4 = (f & 15) << 2;
      float4 v = *reinterpret_cast<const float4*>(
          src + (size_t)(b0 + r) * SIZE + (size_t)kblk * BS + kh * KH + c4);
      *reinterpret_cast<float4*>(&Xs[r][c4]) = v;
    }
    // ---- stage weight half transposed: Wst[c][d] = w[kblk][kh*64+d][c]
    #pragma unroll
    for (int i = 0; i < 8; ++i) {
      int f  = tid + i * 256;
      int d  = f >> 5;                        // 0..63 (reduction dim within half)
      int c4 = (f & 31) << 2;                 // output column group (coalesced read)
      float4 v = *reinterpret_cast<const float4*>(
          wblk + (size_t)(kh * KH + d) * BS + c4);
      Wst[c4 + 0][d] = v.x;
      Wst[c4 + 1][d] = v.y;
      Wst[c4 + 2][d] = v.z;
      Wst[c4 + 3][d] = v.w;
    }
    __syncthreads();

    // ---- 16 K-steps of 4, 4 wmma each (A-frag hoisted over the N-tiles)
    #pragma unroll 4
    for (int kk = 0; kk < KH; kk += 4) {
      v2f a = *reinterpret_cast<const v2f*>(&Xs[mrow][kk + khalf]);
      #pragma unroll
      for (int j = 0; j < 4; ++j) {
        v2f b = *reinterpret_cast<const v2f*>(&Wst[nbase + j * 16 + ncol][kk + khalf]);
        acc[j] = __builtin_amdgcn_wmma_f32_16x16x4_f32(
            /*neg_a=*/false, a, /*neg_b=*/false, b,
            /*c_mod=*/(short)0, acc[j], /*reuse_a=*/false, /*reuse_b=*/false);
      }
    }
  }

  // ---- epilogue: permute folded into store addressing
  const int mtop = (wv & 3) * 16 + ((lane >> 4) << 3);  // M of vgpr r is mtop + r
  #pragma unroll
  for (int j = 0; j < 4; ++j) {
    const int e = nbase + j * 16 + ncol;     // this lane's output column of the block
    size_t col;
    float badd = 0.0f;
    if (PASS == 1) {
      col = (size_t)e * NB + kblk;                              // permute 1: p = c*64 + k
    } else {
      col = (size_t)(e & 63) * BS + 2 * kblk + (e >> 6);        // permute 2
      badd = bias[col];
    }
    float* __restrict__ drow = dst + (size_t)(b0 + mtop) * SIZE + col;
    #pragma unroll
    for (int r = 0; r < 8; ++r) {
      float v = acc[j][r];
      if (PASS == 2) v += badd;
      drow[(size_t)r * SIZE] = v;
    }
  }
}

extern "C" void kernel_launch(void* const* d_in, const int* in_sizes, int n_in,
                              void* d_out, int out_size, void* d_ws, size_t ws_size,
                              hipStream_t stream) {
  const float* x    = (const float*)d_in[0];   // [NTOK, SIZE]
  const float* w1   = (const float*)d_in[1];   // [NB, BS, BS]
  const float* w2   = (const float*)d_in[2];   // [NB, BS, BS]
  const float* bias = (const float*)d_in[3];   // [SIZE]
  float* out   = (float*)d_out;                // [NTOK, SIZE]
  float* inter = (float*)d_ws;                 // needs NTOK*SIZE*4 = 134 MB scratch

  const int Btok = in_sizes[0] / SIZE;         // 4096
  dim3 block(256);
  dim3 grid(Btok / TM, NB);

  monarch_pass<1><<<grid, block, 0, stream>>>(x,     w1, bias, inter);
  monarch_pass<2><<<grid, block, 0, stream>>>(inter, w2, bias, out);
}